// AngularPenaltySMLossWithSoftLabel_8993661517957
// MI455X (gfx1250) — compile-verified
//
#include <hip/hip_runtime.h>
#include <cstdint>

// ---------------------------------------------------------------- constants
#define NROWS 2048
#define DDIM  512
#define CDIM  10000

#define SSCALE 64.0f
#define EPSF   1e-7f
#define KDECAY 0.2f
#define COSM   0.8775825618903728f   // cos(0.5)
#define SINM   0.4794255386042030f   // sin(0.5)

#define KC      32                   // K chunk
#define NCH     (DDIM / KC)          // 16 chunks
#define GCOLS   128                  // columns per group (8 waves x 16)
#define NGROUPS ((CDIM + GCOLS - 1) / GCOLS)   // 79
#define NSEG    4
#define ASTRIDE 36                   // 32 + 4 pad floats (16B-aligned row stride)
#define WSTRIDE 36

typedef float v2f __attribute__((ext_vector_type(2)));
typedef float v8f __attribute__((ext_vector_type(8)));

#if defined(__HIP_DEVICE_COMPILE__) && __has_builtin(__builtin_amdgcn_tensor_load_to_lds)
#define HAVE_TDM 1
#else
#define HAVE_TDM 0
#endif

#if HAVE_TDM
typedef unsigned int u32x4 __attribute__((ext_vector_type(4)));
typedef int          i32x8 __attribute__((ext_vector_type(8)));
typedef int          i32x4 __attribute__((ext_vector_type(4)));

// D# group-1 for a 2D fp32 tile (KC x tile_d1) with LDS row padding:
// pad_interval=4 -> every 32 DWORDs, pad_amount=3 -> +4 DWORDs (stride 36).
__device__ __forceinline__ i32x8 tdm_make_g1(unsigned tensor_d1, unsigned tile_d1) {
  i32x8 g1;
  g1[0] = (int)((2u << 16) | (1u << 20) | (4u << 22) | (3u << 25)); // 4B elems, pad on
  g1[1] = (int)((unsigned)KC << 16);                                 // tensor_dim0 = KC
  g1[2] = (int)((tensor_d1 & 0xFFFFu) << 16);                        // tensor_dim1 lo
  g1[3] = (int)(((tensor_d1 >> 16) & 0xFFFFu) | ((unsigned)KC << 16)); // td1 hi | tile_d0
  g1[4] = (int)(tile_d1 & 0xFFFFu);                                  // tile_dim1, td2=0
  g1[5] = (int)DDIM;                                                 // dim0 stride lo
  g1[6] = 0;
  g1[7] = 0;
  return g1;
}

// Issue one TDM load given precomputed group-1; only group-0 varies per chunk.
__device__ __forceinline__ void tdm_issue(unsigned long long ga, unsigned lds_off,
                                          const i32x8& g1) {
  u32x4 g0;
  g0.x = 1u;                                               // count=1, user mode
  g0.y = lds_off;                                          // lds_addr (bytes)
  g0.z = (unsigned)(ga & 0xFFFFFFFFu);                     // global_addr[31:0]
  g0.w = (unsigned)((ga >> 32) & 0x1FFFFFFu) | (2u << 30); // addr[56:32] | type=2
  i32x4 z4 = {0, 0, 0, 0};
  i32x8 z8 = {0, 0, 0, 0, 0, 0, 0, 0};
  __builtin_amdgcn_tensor_load_to_lds(g0, g1, z4, z4, z8, 0);
}
#endif

template <int NN>
__device__ __forceinline__ void wait_tensorcnt() {
#if HAVE_TDM && __has_builtin(__builtin_amdgcn_s_wait_tensorcnt)
  __builtin_amdgcn_s_wait_tensorcnt((short)NN);
#endif
}

// ---------------------------------------------------------------- kernel 1
__global__ __launch_bounds__(256) void rownorm_kernel(const float* __restrict__ x,
                                                      float* __restrict__ inv) {
  const int row = blockIdx.x;
  const float* xr = x + (size_t)row * DDIM;
  float ss = 0.f;
  for (int i = threadIdx.x; i < DDIM; i += 256) { float v = xr[i]; ss += v * v; }
  for (int m = 16; m >= 1; m >>= 1) ss += __shfl_xor(ss, m, 32);
  __shared__ float part[8];
  if ((threadIdx.x & 31) == 0) part[threadIdx.x >> 5] = ss;
  __syncthreads();
  if (threadIdx.x == 0) {
    float t = 0.f;
    for (int i = 0; i < 8; ++i) t += part[i];
    float n = fmaxf(sqrtf(t), 1e-12f);
    inv[row] = 1.0f / n;
  }
}

// ---------------------------------------------------------------- kernel 2
// grid (128 strips, 4 col-segments), 256 threads = 8 waves.
__global__ __launch_bounds__(256) void arcface_main_kernel(
    const float* __restrict__ x, const int* __restrict__ labels,
    const float* __restrict__ Wm, const float* __restrict__ inv,
    float* __restrict__ wf, float* __restrict__ seG, float* __restrict__ lsG,
    float* __restrict__ ssG, float* __restrict__ wfyG) {
  __shared__ float As[2][16 * ASTRIDE];
  __shared__ float Ws[2][GCOLS * WSTRIDE];
  __shared__ float rAcc[3][16];

  const int strip = blockIdx.x;
  const int seg   = blockIdx.y;
  const int tid   = threadIdx.x;
  const int lane  = tid & 31;
  const int wave  = tid >> 5;
  const int half  = lane >> 4;   // 0 or 1 (K sub-pair / M+8 select)
  const int l16   = lane & 15;

  if (tid < 16) { rAcc[0][tid] = 0.f; rAcc[1][tid] = 0.f; rAcc[2][tid] = 0.f; }

  // per-lane metadata for its 8 output rows: row = strip*16 + 8*half + j
  float invr[8]; int ylab[8];
#pragma unroll
  for (int j = 0; j < 8; ++j) {
    int r = strip * 16 + 8 * half + j;
    invr[j] = inv[r];
    ylab[j] = labels[r];
  }

  float se[8], ls[8], ssm[8];
#pragma unroll
  for (int j = 0; j < 8; ++j) { se[j] = 0.f; ls[j] = 0.f; ssm[j] = 0.f; }

  const float* xbase = x + (size_t)strip * 16 * DDIM;
  // lane-local LDS read offsets (floats); + kk*4 per k-step
  const int aoff = l16 * ASTRIDE + 2 * half;
  const int boff = (wave * 16 + l16) * WSTRIDE + 2 * half;

#if HAVE_TDM
  const unsigned asOff[2] = {(unsigned)(size_t)(void*)&As[0][0],
                             (unsigned)(size_t)(void*)&As[1][0]};
  const unsigned wsOff[2] = {(unsigned)(size_t)(void*)&Ws[0][0],
                             (unsigned)(size_t)(void*)&Ws[1][0]};
  const unsigned long long gaA0 = (unsigned long long)(uintptr_t)xbase;
  const i32x8 g1A = tdm_make_g1(16u, 16u);            // fully loop-invariant
#endif

  v8f acc;
  for (int g = seg; g < NGROUPS; g += NSEG) {
    const int colBase = g * GCOLS;
#pragma unroll
    for (int q = 0; q < 8; ++q) acc[q] = 0.f;

#if HAVE_TDM
    const unsigned long long gaW0 =
        (unsigned long long)(uintptr_t)(Wm + (size_t)colBase * DDIM);
    const i32x8 g1W = tdm_make_g1((unsigned)(CDIM - colBase), GCOLS); // per group
    if (wave == 0) {
      tdm_issue(gaW0, wsOff[0], g1W);
      tdm_issue(gaA0, asOff[0], g1A);
    }
#else
    auto stage_fb = [&](int c, int buf) {
      int kb = c * KC;
      {
        int row = tid >> 4, kk = (tid & 15) * 2;
        const float* gp = xbase + row * DDIM + kb + kk;
        As[buf][row * ASTRIDE + kk]     = gp[0];
        As[buf][row * ASTRIDE + kk + 1] = gp[1];
      }
#pragma unroll
      for (int i = 0; i < 4; ++i) {
        int idx = tid + i * 256;
        int col = idx >> 3, q = idx & 7;
        int gcol = colBase + col; if (gcol >= CDIM) gcol = CDIM - 1;
        const float4 v = *(const float4*)(Wm + (size_t)gcol * DDIM + kb + q * 4);
        *(float4*)&Ws[buf][col * WSTRIDE + q * 4] = v;
      }
    };
    stage_fb(0, 0);
#endif

    for (int c = 0; c < NCH; ++c) {
      const int buf = c & 1;
#if HAVE_TDM
      if (wave == 0) {
        if (c + 1 < NCH) {
          const unsigned kbB = (unsigned)((c + 1) * KC) * 4u;  // byte offset
          tdm_issue(gaW0 + kbB, wsOff[buf ^ 1], g1W);
          tdm_issue(gaA0 + kbB, asOff[buf ^ 1], g1A);
          wait_tensorcnt<2>();
        } else {
          wait_tensorcnt<0>();
        }
      }
#else
      if (c + 1 < NCH) stage_fb(c + 1, buf ^ 1);
#endif
      __syncthreads();
      const float* Ap = &As[buf][aoff];
      const float* Bp = &Ws[buf][boff];
#pragma unroll
      for (int kk = 0; kk < 8; ++kk) {
        v2f a = *(const v2f*)(Ap + kk * 4);
        v2f b = *(const v2f*)(Bp + kk * 4);
        acc = __builtin_amdgcn_wmma_f32_16x16x4_f32(
            false, a, false, b, (short)0, acc, false, false);
      }
      __syncthreads();
    }

    // epilogue for this 16x16 tile (wave-uniform validity: CDIM % 16 == 0)
    const int col = colBase + wave * 16 + l16;
    if (colBase + wave * 16 < CDIM) {
      const int rowb = strip * 16 + 8 * half;
#pragma unroll
      for (int j = 0; j < 8; ++j) {
        const int row = rowb + j;
        const float wfv = acc[j] * invr[j];
        wf[(size_t)row * CDIM + col] = wfv;
        float t   = fminf(fmaxf(wfv, -1.0f + EPSF), 1.0f - EPSF);
        float num = SSCALE * (t * COSM - SINM * sqrtf(fmaxf(1.0f - t * t, 0.0f)));
        se[j] += __expf(SSCALE * wfv);
        int d = col - ylab[j];
        if (d > -(CDIM / 2) && d <= (CDIM / 2)) {
          float w = __expf(-KDECAY * fabsf((float)d));
          ls[j]  += w * num;
          ssm[j] += w;
          if (d == 0) wfyG[row] = wfv;   // written by exactly one lane globally
        }
      }
    }
  }

  // reduce 16 lanes (same row) -> lane l16==0, then combine 8 waves in LDS
#pragma unroll
  for (int j = 0; j < 8; ++j) {
    float a = se[j], b = ls[j], c = ssm[j];
    for (int m = 8; m >= 1; m >>= 1) {
      a += __shfl_xor(a, m, 32);
      b += __shfl_xor(b, m, 32);
      c += __shfl_xor(c, m, 32);
    }
    if (l16 == 0) {
      int r = 8 * half + j;
      atomicAdd(&rAcc[0][r], a);
      atomicAdd(&rAcc[1][r], b);
      atomicAdd(&rAcc[2][r], c);
    }
  }
  __syncthreads();
  if (tid < 16) {
    const int row = strip * 16 + tid;
    seG[seg * NROWS + row] = rAcc[0][tid];
    lsG[seg * NROWS + row] = rAcc[1][tid];
    ssG[seg * NROWS + row] = rAcc[2][tid];
  }
}

// ---------------------------------------------------------------- kernel 3
__global__ __launch_bounds__(256) void rowfinal_kernel(
    const float* __restrict__ seG, const float* __restrict__ lsG,
    const float* __restrict__ ssG, const float* __restrict__ wfyG,
    float* __restrict__ rowL) {
  const int row = blockIdx.x * 256 + threadIdx.x;
  if (row >= NROWS) return;
  float se = 0.f, ls = 0.f, ss = 0.f;
#pragma unroll
  for (int s = 0; s < NSEG; ++s) {
    se += seG[s * NROWS + row];
    ls += lsG[s * NROWS + row];
    ss += ssG[s * NROWS + row];
  }
  const float wfy = wfyG[row];
  float t    = fminf(fmaxf(wfy, -1.0f + EPSF), 1.0f - EPSF);
  float numy = SSCALE * (t * COSM - SINM * sqrtf(fmaxf(1.0f - t * t, 0.0f)));
  float den  = __expf(numy) + se - __expf(SSCALE * wfy);
  rowL[row]  = ls / ss - __logf(den);
}

// ---------------------------------------------------------------- kernel 4
__global__ __launch_bounds__(256) void lossreduce_kernel(
    const float* __restrict__ rowL, float* __restrict__ out) {
  float s = 0.f;
  for (int i = threadIdx.x; i < NROWS; i += 256) s += rowL[i];
  for (int m = 16; m >= 1; m >>= 1) s += __shfl_xor(s, m, 32);
  __shared__ float part[8];
  if ((threadIdx.x & 31) == 0) part[threadIdx.x >> 5] = s;
  __syncthreads();
  if (threadIdx.x == 0) {
    float t = 0.f;
    for (int i = 0; i < 8; ++i) t += part[i];
    out[(size_t)NROWS * CDIM] = -t / (float)NROWS;
  }
}

// ---------------------------------------------------------------- launch
extern "C" void kernel_launch(void* const* d_in, const int* in_sizes, int n_in,
                              void* d_out, int out_size, void* d_ws, size_t ws_size,
                              hipStream_t stream) {
  const float* x      = (const float*)d_in[0];
  const int*   labels = (const int*)d_in[1];
  const float* Wm     = (const float*)d_in[2];
  float* out = (float*)d_out;          // [0, NROWS*CDIM) = wf ; [NROWS*CDIM] = loss

  float* inv  = (float*)d_ws;          // NROWS
  float* seG  = inv  + NROWS;          // NSEG*NROWS
  float* lsG  = seG  + NSEG * NROWS;   // NSEG*NROWS
  float* ssG  = lsG  + NSEG * NROWS;   // NSEG*NROWS
  float* wfyG = ssG  + NSEG * NROWS;   // NROWS
  float* rowL = wfyG + NROWS;          // NROWS

  rownorm_kernel<<<NROWS, 256, 0, stream>>>(x, inv);
  arcface_main_kernel<<<dim3(NROWS / 16, NSEG), 256, 0, stream>>>(
      x, labels, Wm, inv, out, seG, lsG, ssG, wfyG);
  rowfinal_kernel<<<NROWS / 256, 256, 0, stream>>>(seG, lsG, ssG, wfyG, rowL);
  lossreduce_kernel<<<1, 256, 0, stream>>>(rowL, out);
}